// ConversationAwareGATLayer_19413252178002
// MI455X (gfx1250) — compile-verified
//
#include <hip/hip_runtime.h>
#include <hip/hip_bf16.h>

// Problem constants (match reference)
#define NN   50000
#define EE   800000
#define INF_ 128
#define HH   4
#define DD   32
#define NEG_SLOPE 0.2f

typedef float v2f __attribute__((ext_vector_type(2)));
typedef float v8f __attribute__((ext_vector_type(8)));

// ---------------------------------------------------------------------------
// Init accumulators, float4-vectorized. Covers:
//   emax[N*H] = -inf, denom[N*H] = 0, rst[N*128] = 0
// ---------------------------------------------------------------------------
__global__ void init_accum(float4* __restrict__ emax, float4* __restrict__ denom,
                           float4* __restrict__ rst) {
  int i = blockIdx.x * blockDim.x + threadIdx.x;   // float4 index
  const float ninf = -__builtin_inff();
  if (i < NN * HH / 4) {
    emax[i]  = make_float4(ninf, ninf, ninf, ninf);
    denom[i] = make_float4(0.f, 0.f, 0.f, 0.f);
  }
  if (i < NN * INF_ / 4) rst[i] = make_float4(0.f, 0.f, 0.f, 0.f);
}

// ---------------------------------------------------------------------------
// GEMM (out = A[Mx128] @ W[128x128]), fp32 WMMA 16x16x4.
// One wave = one 16x16 tile; block = 8 waves = one 16-row strip.
// ---------------------------------------------------------------------------
__global__ __launch_bounds__(256) void gemm_feat(const float* __restrict__ A,
                                                 const float* __restrict__ W,
                                                 float* __restrict__ out) {
  const int lane = threadIdx.x & 31;
  const int wave = threadIdx.x >> 5;
  const int r0 = blockIdx.x * 16;
  const int c0 = wave * 16;
  const int m = lane & 15;
  const int koff = (lane >> 4) << 1;  // 0 or 2
  const float* __restrict__ arow = A + (r0 + m) * INF_;
  v8f acc = {};
  for (int k = 0; k < INF_; k += 4) {
    const int kk = k + koff;
    v2f a; a.x = arow[kk];                 a.y = arow[kk + 1];
    v2f b; b.x = W[kk * INF_ + c0 + m];    b.y = W[(kk + 1) * INF_ + c0 + m];
    acc = __builtin_amdgcn_wmma_f32_16x16x4_f32(false, a, false, b,
                                                (short)0, acc, false, false);
  }
  const int rbase = r0 + ((lane >> 4) << 3);
  float* __restrict__ ocol = out + c0 + m;
#pragma unroll
  for (int r = 0; r < 8; ++r)
    ocol[(rbase + r) * INF_] = acc[r];
}

// ---------------------------------------------------------------------------
// el/er: per (node, head) dot with attn_l / attn_r (float4 inner loop)
// ---------------------------------------------------------------------------
__global__ void attn_scores(const float* __restrict__ feat,
                            const float* __restrict__ attn_l,
                            const float* __restrict__ attn_r,
                            float* __restrict__ el, float* __restrict__ er) {
  int i = blockIdx.x * blockDim.x + threadIdx.x;
  if (i >= NN * HH) return;
  const int n = i >> 2, h = i & 3;
  const float4* __restrict__ f  = (const float4*)(feat + n * INF_ + h * DD);
  const float4* __restrict__ al = (const float4*)(attn_l + h * DD);
  const float4* __restrict__ ar = (const float4*)(attn_r + h * DD);
  float sl = 0.0f, sr = 0.0f;
#pragma unroll
  for (int d = 0; d < DD / 4; ++d) {
    float4 v = f[d], a = al[d], b = ar[d];
    sl += v.x * a.x + v.y * a.y + v.z * a.z + v.w * a.w;
    sr += v.x * b.x + v.y * b.y + v.z * b.z + v.w * b.w;
  }
  el[i] = sl;
  er[i] = sr;
}

// ---------------------------------------------------------------------------
// Edge pass A: one thread per EDGE. float4 over the 4 heads.
//   e = leaky_relu(el[src]+er[dst]); atomic float max into emax[dst]
// ---------------------------------------------------------------------------
__global__ void edge_scores(const int* __restrict__ src, const int* __restrict__ dst,
                            const float4* __restrict__ el4, const float4* __restrict__ er4,
                            float4* __restrict__ ebuf4, float* __restrict__ emax) {
  int e = blockIdx.x * blockDim.x + threadIdx.x;
  if (e >= EE) return;
  const int s = src[e], d = dst[e];
  const float4 l = el4[s];
  const float4 r = er4[d];
  float4 v;
  v.x = l.x + r.x; v.x = v.x > 0.f ? v.x : NEG_SLOPE * v.x;
  v.y = l.y + r.y; v.y = v.y > 0.f ? v.y : NEG_SLOPE * v.y;
  v.z = l.z + r.z; v.z = v.z > 0.f ? v.z : NEG_SLOPE * v.z;
  v.w = l.w + r.w; v.w = v.w > 0.f ? v.w : NEG_SLOPE * v.w;
  ebuf4[e] = v;
  float* em = emax + d * HH;
  atomicMax(em + 0, v.x);
  atomicMax(em + 1, v.y);
  atomicMax(em + 2, v.z);
  atomicMax(em + 3, v.w);
}

// ---------------------------------------------------------------------------
// Edge pass B: one thread per EDGE. a = exp(e - emax[dst]); atomicAdd denom.
// ---------------------------------------------------------------------------
__global__ void edge_exp(const int* __restrict__ dst,
                         const float4* __restrict__ emax4,
                         float4* __restrict__ ebuf4, float* __restrict__ denom) {
  int e = blockIdx.x * blockDim.x + threadIdx.x;
  if (e >= EE) return;
  const int d = dst[e];
  const float4 mx = emax4[d];
  float4 v = ebuf4[e];
  v.x = __expf(v.x - mx.x);
  v.y = __expf(v.y - mx.y);
  v.z = __expf(v.z - mx.z);
  v.w = __expf(v.w - mx.w);
  ebuf4[e] = v;
  float* dn = denom + d * HH;
  atomicAdd(dn + 0, v.x);
  atomicAdd(dn + 1, v.y);
  atomicAdd(dn + 2, v.z);
  atomicAdd(dn + 3, v.w);
}

// ---------------------------------------------------------------------------
// Edge pass C: ONE WAVE (32 lanes) per edge, float4 per lane (b128 gathers).
//   lane t handles features [t*4, t*4+4); head = t>>3.
//   rst[dst] += feat[src] * alpha
// ---------------------------------------------------------------------------
__global__ void edge_aggregate(const int* __restrict__ src, const int* __restrict__ dst,
                               const float4* __restrict__ feat4,
                               const float* __restrict__ ebuf,
                               const float* __restrict__ denom,
                               float* __restrict__ rst) {
  long long idx = (long long)blockIdx.x * blockDim.x + threadIdx.x;
  const int e = (int)(idx >> 5);
  const int t = (int)(idx & 31);          // float4 slot within the row
  if (e >= EE) return;
  const int s = src[e], d = dst[e];
  const int h = t >> 3;                   // 8 float4 slots per head
  const float alpha = ebuf[e * HH + h] / denom[d * HH + h];
  const float4 f = feat4[s * (INF_ / 4) + t];
  float* r = rst + d * INF_ + t * 4;
  atomicAdd(r + 0, f.x * alpha);
  atomicAdd(r + 1, f.y * alpha);
  atomicAdd(r + 2, f.z * alpha);
  atomicAdd(r + 3, f.w * alpha);
}

// ---------------------------------------------------------------------------
// h_gat = elu(rst + bias_gat), float4-vectorized (into feat's buffer)
// ---------------------------------------------------------------------------
__global__ void elu_bias(const float4* __restrict__ rst4,
                         const float4* __restrict__ bias4,
                         float4* __restrict__ h_gat4) {
  int i = blockIdx.x * blockDim.x + threadIdx.x;
  if (i >= NN * INF_ / 4) return;
  float4 x = rst4[i];
  const float4 b = bias4[i & (INF_ / 4 - 1)];
  x.x += b.x; x.y += b.y; x.z += b.z; x.w += b.w;
  x.x = x.x > 0.f ? x.x : (__expf(x.x) - 1.f);
  x.y = x.y > 0.f ? x.y : (__expf(x.y) - 1.f);
  x.z = x.z > 0.f ? x.z : (__expf(x.z) - 1.f);
  x.w = x.w > 0.f ? x.w : (__expf(x.w) - 1.f);
  h_gat4[i] = x;
}

// ---------------------------------------------------------------------------
// GEMM + bias: h_proj = h_gat @ W_proj + b_proj   (into rst's buffer)
// ---------------------------------------------------------------------------
__global__ __launch_bounds__(256) void gemm_proj(const float* __restrict__ A,
                                                 const float* __restrict__ W,
                                                 const float* __restrict__ bias,
                                                 float* __restrict__ out) {
  const int lane = threadIdx.x & 31;
  const int wave = threadIdx.x >> 5;
  const int r0 = blockIdx.x * 16;
  const int c0 = wave * 16;
  const int m = lane & 15;
  const int koff = (lane >> 4) << 1;
  const float* __restrict__ arow = A + (r0 + m) * INF_;
  v8f acc = {};
  for (int k = 0; k < INF_; k += 4) {
    const int kk = k + koff;
    v2f a; a.x = arow[kk];              a.y = arow[kk + 1];
    v2f b; b.x = W[kk * INF_ + c0 + m]; b.y = W[(kk + 1) * INF_ + c0 + m];
    acc = __builtin_amdgcn_wmma_f32_16x16x4_f32(false, a, false, b,
                                                (short)0, acc, false, false);
  }
  const int rbase = r0 + ((lane >> 4) << 3);
  const int col = c0 + m;
  const float bc = bias[col];
#pragma unroll
  for (int r = 0; r < 8; ++r)
    out[(rbase + r) * INF_ + col] = acc[r] + bc;
}

// ---------------------------------------------------------------------------
// GEMM over K=256 split-A (h_proj | ctx) + fused sigmoid gate + blend:
//   out = g * h_proj + (1-g) * ctx,  g = sigmoid([h_proj|ctx] @ W_gate + b)
// ---------------------------------------------------------------------------
__global__ __launch_bounds__(256) void gemm_gate(const float* __restrict__ hp,
                                                 const float* __restrict__ ctx,
                                                 const float* __restrict__ W,
                                                 const float* __restrict__ bias,
                                                 float* __restrict__ out) {
  const int lane = threadIdx.x & 31;
  const int wave = threadIdx.x >> 5;
  const int r0 = blockIdx.x * 16;
  const int c0 = wave * 16;
  const int m = lane & 15;
  const int koff = (lane >> 4) << 1;
  const float* __restrict__ a0 = hp  + (r0 + m) * INF_;
  const float* __restrict__ a1 = ctx + (r0 + m) * INF_;
  v8f acc = {};
  for (int k = 0; k < 2 * INF_; k += 4) {
    const int kk = k + koff;
    // kk is even; (kk,kk+1) never straddles the 128 boundary
    v2f a;
    if (kk < INF_) { a.x = a0[kk]; a.y = a0[kk + 1]; }
    else           { a.x = a1[kk - INF_]; a.y = a1[kk + 1 - INF_]; }
    v2f b; b.x = W[kk * INF_ + c0 + m]; b.y = W[(kk + 1) * INF_ + c0 + m];
    acc = __builtin_amdgcn_wmma_f32_16x16x4_f32(false, a, false, b,
                                                (short)0, acc, false, false);
  }
  const int rbase = r0 + ((lane >> 4) << 3);
  const int col = c0 + m;
  const float bc = bias[col];
#pragma unroll
  for (int r = 0; r < 8; ++r) {
    const int row = rbase + r;
    const float g = 1.0f / (1.0f + __expf(-(acc[r] + bc)));
    const float hv = hp[row * INF_ + col];
    const float cv = ctx[row * INF_ + col];
    out[row * INF_ + col] = g * hv + (1.0f - g) * cv;
  }
}

// ---------------------------------------------------------------------------
// Launch
// ---------------------------------------------------------------------------
extern "C" void kernel_launch(void* const* d_in, const int* in_sizes, int n_in,
                              void* d_out, int out_size, void* d_ws, size_t ws_size,
                              hipStream_t stream) {
  const float* h      = (const float*)d_in[0];
  const int*   src    = (const int*)  d_in[1];
  const int*   dst    = (const int*)  d_in[2];
  const float* ctx    = (const float*)d_in[3];
  const float* W_fc   = (const float*)d_in[4];
  const float* attn_l = (const float*)d_in[5];
  const float* attn_r = (const float*)d_in[6];
  const float* b_gat  = (const float*)d_in[7];
  const float* W_proj = (const float*)d_in[8];
  const float* b_proj = (const float*)d_in[9];
  const float* W_gate = (const float*)d_in[10];
  const float* b_gate = (const float*)d_in[11];
  float* out = (float*)d_out;

  // Workspace layout (floats). feat doubles as h_gat; rst doubles as h_proj.
  float* ws    = (float*)d_ws;
  float* feat  = ws;                       // N*128   (later: h_gat)
  float* el    = feat  + (size_t)NN * INF_;// N*H
  float* er    = el    + (size_t)NN * HH;  // N*H
  float* emax  = er    + (size_t)NN * HH;  // N*H
  float* denom = emax  + (size_t)NN * HH;  // N*H
  float* ebuf  = denom + (size_t)NN * HH;  // E*H
  float* rst   = ebuf  + (size_t)EE * HH;  // N*128  (later: h_proj)

  const int TB = 256;

  // 1) init accumulators (float4 granularity)
  init_accum<<<(NN * INF_ / 4 + TB - 1) / TB, TB, 0, stream>>>(
      (float4*)emax, (float4*)denom, (float4*)rst);
  // 2) feat = h @ W_fc (WMMA fp32)
  gemm_feat<<<NN / 16, TB, 0, stream>>>(h, W_fc, feat);
  // 3) el / er
  attn_scores<<<(NN * HH + TB - 1) / TB, TB, 0, stream>>>(feat, attn_l, attn_r, el, er);
  // 4) edge scores + segment max (thread per edge)
  edge_scores<<<(EE + TB - 1) / TB, TB, 0, stream>>>(
      src, dst, (const float4*)el, (const float4*)er, (float4*)ebuf, emax);
  // 5) exp + segment sum (thread per edge)
  edge_exp<<<(EE + TB - 1) / TB, TB, 0, stream>>>(
      dst, (const float4*)emax, (float4*)ebuf, denom);
  // 6) weighted scatter-add aggregation (wave per edge, float4 per lane)
  edge_aggregate<<<(int)(((long long)EE * 32 + TB - 1) / TB), TB, 0, stream>>>(
      src, dst, (const float4*)feat, ebuf, denom, rst);
  // 7) h_gat = elu(rst + bias)  (into feat buffer)
  elu_bias<<<(NN * INF_ / 4 + TB - 1) / TB, TB, 0, stream>>>(
      (const float4*)rst, (const float4*)b_gat, (float4*)feat);
  // 8) h_proj = h_gat @ W_proj + b_proj  (into rst buffer)
  gemm_proj<<<NN / 16, TB, 0, stream>>>(feat, W_proj, b_proj, rst);
  // 9) gate GEMM + sigmoid + blend -> out
  gemm_gate<<<NN / 16, TB, 0, stream>>>(rst, ctx, W_gate, b_gate, out);
}